// CAMInjector_69750268887295
// MI455X (gfx1250) — compile-verified
//
#include <hip/hip_runtime.h>
#include <hip/hip_bf16.h>
#include <math.h>

#define BATCH 4
#define SEQ   4096
#define DK    64
#define NMEM  4
#define RANK  64
#define STILES (SEQ / 16)   // 256 tiles per dim
#define INV_SQRT_DK 0.125f

typedef __attribute__((ext_vector_type(2))) float v2f;
typedef __attribute__((ext_vector_type(8))) float v8f;

// ---------------------------------------------------------------------------
// Kernel 1: Q_summary[b][d] = mean over seq of Q[b,s,d]
// ---------------------------------------------------------------------------
__global__ void qsum_kernel(const float* __restrict__ Q, float* __restrict__ qs) {
    const int b = blockIdx.x;
    const int t = threadIdx.x;          // 256 threads
    const int d = t & 63;
    const int chunk = t >> 6;           // 0..3
    const float* base = Q + (size_t)b * SEQ * DK;
    float acc = 0.0f;
    for (int s = chunk; s < SEQ; s += 4)
        acc += base[(size_t)s * DK + d];
    __shared__ float red[256];
    red[t] = acc;
    __syncthreads();
    if (t < 64) {
        float v = red[t] + red[t + 64] + red[t + 128] + red[t + 192];
        qs[b * DK + t] = v * (1.0f / (float)SEQ);
    }
}

// ---------------------------------------------------------------------------
// Kernel 2: all the small math (pv, MLP, softmax/entropy, MV, MVw, ||MVw||)
// one block, 256 threads: thread t -> (b = t>>6, d = t&63)
// ---------------------------------------------------------------------------
__global__ void small_kernel(const float* __restrict__ mpv,
                             const float* __restrict__ qs,
                             const float* __restrict__ w1,
                             const float* __restrict__ b1,
                             const float* __restrict__ w2,
                             const float* __restrict__ b2,
                             const float* __restrict__ wmem,
                             float* __restrict__ mv_out,
                             float* __restrict__ mvw_out,
                             float* __restrict__ norm_out,
                             float* __restrict__ entropy_out) {
    const int t = threadIdx.x;
    const int b = t >> 6;
    const int d = t & 63;

    __shared__ float s_pv[NMEM][BATCH][DK];
    __shared__ float s_h[NMEM][BATCH][DK];
    __shared__ float s_scores[NMEM][BATCH];
    __shared__ float s_alpha[BATCH][NMEM];
    __shared__ float s_mv[BATCH][DK];
    __shared__ float s_sq[BATCH][DK];

    // pv[n,b,d] = mean_r mpv[NMEM-1-n, b, r, d] * gamma^(n+1)
    float decay = 1.0f;
    for (int n = 0; n < NMEM; n++) {
        decay *= 0.9f;
        const float* p = mpv + (((size_t)(NMEM - 1 - n) * BATCH + b) * RANK) * DK + d;
        float acc = 0.0f;
        for (int r = 0; r < RANK; r++) acc += p[(size_t)r * DK];
        s_pv[n][b][d] = acc * (1.0f / (float)RANK) * decay;
    }
    __syncthreads();

    // h[n,b,j] = relu(sum_e in[e]*w1[j,e] + b1[j]);  in = [qs[b], pv[n,b]]
    for (int n = 0; n < NMEM; n++) {
        float acc = b1[d];
        const float* wr = w1 + (size_t)d * 128;
        for (int e = 0; e < 64; e++) acc += qs[b * DK + e] * wr[e];
        for (int e = 0; e < 64; e++) acc += s_pv[n][b][e] * wr[64 + e];
        s_h[n][b][d] = fmaxf(acc, 0.0f);
    }
    __syncthreads();

    // scores[n][b]
    if (d < NMEM) {
        const int n = d;
        float acc = b2[0];
        for (int j = 0; j < 64; j++) acc += s_h[n][b][j] * w2[j];
        s_scores[n][b] = acc;
    }
    __syncthreads();

    // softmax over n + entropy
    if (d == 0) {
        float mx = s_scores[0][b];
        for (int n = 1; n < NMEM; n++) mx = fmaxf(mx, s_scores[n][b]);
        float ex[NMEM], den = 0.0f;
        for (int n = 0; n < NMEM; n++) { ex[n] = expf(s_scores[n][b] - mx); den += ex[n]; }
        float ent = 0.0f;
        for (int n = 0; n < NMEM; n++) {
            float a = ex[n] / den;
            s_alpha[b][n] = a;
            ent -= a * logf(a + 1e-9f);
        }
        entropy_out[b] = ent;
    }
    __syncthreads();

    // MV_summary[b,d]
    float mv = 0.0f;
    for (int n = 0; n < NMEM; n++) mv += s_pv[n][b][d] * s_alpha[b][n];
    s_mv[b][d] = mv;
    mv_out[b * DK + d] = mv;
    __syncthreads();

    // MVw[b,d] = sum_e MV[b,e] * W[d,e]
    float mvw = 0.0f;
    const float* wr = wmem + (size_t)d * DK;
    for (int e = 0; e < DK; e++) mvw += s_mv[b][e] * wr[e];
    mvw_out[b * DK + d] = mvw;
    s_sq[b][d] = mvw * mvw;
    __syncthreads();

    if (d == 0) {
        float acc = 0.0f;
        for (int e = 0; e < DK; e++) acc += s_sq[b][e];
        norm_out[b] = sqrtf(acc);
    }
}

// ---------------------------------------------------------------------------
// Kernel 3: G[b,s] = sigmoid(gg * (Q . MV)/sqrt(dk) + gb),  u[b,t] = (K . MVw)/sqrt(dk)
// blockIdx: 0..3 -> G for b, 4..7 -> u for b.  Also gate stats to d_out.
// ---------------------------------------------------------------------------
__global__ void gu_kernel(const float* __restrict__ Q,
                          const float* __restrict__ K,
                          const float* __restrict__ mv,
                          const float* __restrict__ mvw,
                          const float* __restrict__ norm,
                          const float* __restrict__ g_gamma,
                          const float* __restrict__ g_bias,
                          float* __restrict__ G_out,
                          float* __restrict__ u_out,
                          float* __restrict__ stats_out) {
    const int b = blockIdx.x & 3;
    const int which = blockIdx.x >> 2;   // 0 = G, 1 = u
    const int t = threadIdx.x;

    __shared__ float vec[DK];
    if (t < DK) vec[t] = which ? mvw[b * DK + t] : mv[b * DK + t];
    __syncthreads();

    const float gg = g_gamma[0];
    const float gb = g_bias[0];
    const float* src = (which ? K : Q) + (size_t)b * SEQ * DK;

    float sumG = 0.0f, sumG2 = 0.0f;
    for (int s = t; s < SEQ; s += 256) {
        const float4* row = (const float4*)(src + (size_t)s * DK);
        float acc = 0.0f;
        for (int i = 0; i < 16; i++) {
            float4 v = row[i];
            acc += v.x * vec[4 * i] + v.y * vec[4 * i + 1] +
                   v.z * vec[4 * i + 2] + v.w * vec[4 * i + 3];
        }
        acc *= INV_SQRT_DK;
        if (which == 0) {
            float g = 1.0f / (1.0f + expf(-(gg * acc + gb)));
            G_out[b * SEQ + s] = g;
            sumG += g; sumG2 += g * g;
        } else {
            u_out[b * SEQ + s] = acc;
        }
    }

    if (which == 0) {
        __shared__ float r1[256], r2[256];
        r1[t] = sumG; r2[t] = sumG2;
        __syncthreads();
        for (int off = 128; off > 0; off >>= 1) {
            if (t < off) { r1[t] += r1[t + off]; r2[t] += r2[t + off]; }
            __syncthreads();
        }
        if (t == 0) {
            float mean = r1[0] / (float)SEQ;
            stats_out[b]      = mean;                                            // gate_impact
            stats_out[4 + b]  = (r2[0] - (float)SEQ * mean * mean) / (float)(SEQ - 1); // gate_variance (ddof=1)
            stats_out[12 + b] = mean * norm[b];                                  // mem_contrib
        }
    }
}

// ---------------------------------------------------------------------------
// Kernel 4 (hot path, store-bound): memory_bias tiles via rank-1 WMMA.
// One wave per 16x16 tile. D = A(16x4) x B(4x16) with only K=0 populated:
//   A[m,0] = G[s0+m]  (f32 A layout: VGPR0 lanes 0-15 = K=0)
//   B[0,n] = u[t0+n]  (f32 B layout: VGPR0 lanes 0-15 = K=0 row)
// so D[m,n] = G[s0+m] * u[t0+n] exactly in f32. Then cndmask -inf for
// causal (col > row) and key-padding (mask==0) and store 8 dwords/lane.
// Tiles strictly above the diagonal store -inf without any compute.
// ---------------------------------------------------------------------------
__global__ void bias_kernel(const float* __restrict__ G,
                            const float* __restrict__ u,
                            const int* __restrict__ amask,
                            float* __restrict__ out) {
    const int tile = blockIdx.x * (blockDim.x >> 5) + (threadIdx.x >> 5);
    const int b  = tile >> 16;            // 65536 tiles per batch
    const int rem = tile & 0xFFFF;
    const int si = rem >> 8;
    const int ti = rem & 255;
    const int s0 = si * 16, t0 = ti * 16;
    const int lane = threadIdx.x & 31;
    const int half = lane >> 4;
    const int l = lane & 15;
    const size_t obase = (size_t)b * SEQ * SEQ;
    const float NEG_INF = -__builtin_inff();

    if (ti > si) {  // fully above diagonal: pure -inf tile, no loads/compute
        for (int v = 0; v < 8; v++) {
            int row = s0 + v + half * 8;
            out[obase + (size_t)row * SEQ + t0 + l] = NEG_INF;
        }
        return;
    }

    const float gval = G[b * SEQ + s0 + l];
    const float uval = u[b * SEQ + t0 + l];
    const bool  pad  = (amask[b * SEQ + t0 + l] == 0);

    v2f a;  a.x = half ? 0.0f : gval;  a.y = 0.0f;
    v2f bm; bm.x = half ? 0.0f : uval; bm.y = 0.0f;
    v8f c = {0.0f, 0.0f, 0.0f, 0.0f, 0.0f, 0.0f, 0.0f, 0.0f};
    c = __builtin_amdgcn_wmma_f32_16x16x4_f32(
            /*neg_a=*/false, a, /*neg_b=*/false, bm,
            /*c_mod=*/(short)0, c, /*reuse_a=*/false, /*reuse_b=*/false);

    const int col = t0 + l;
    for (int v = 0; v < 8; v++) {
        int row = s0 + v + half * 8;
        float val = (pad || (col > row)) ? NEG_INF : c[v];
        out[obase + (size_t)row * SEQ + col] = val;
    }
}

// ---------------------------------------------------------------------------
// Host launch
// ---------------------------------------------------------------------------
extern "C" void kernel_launch(void* const* d_in, const int* in_sizes, int n_in,
                              void* d_out, int out_size, void* d_ws, size_t ws_size,
                              hipStream_t stream) {
    const float* Q     = (const float*)d_in[0];
    const float* K     = (const float*)d_in[1];
    // d_in[2] = V (unused by the reference)
    const float* mpv   = (const float*)d_in[3];
    const int*   amask = (const int*)d_in[4];
    // d_in[5] = layer_depth (unused)
    const float* wmem  = (const float*)d_in[6];
    const float* w1    = (const float*)d_in[7];
    const float* b1    = (const float*)d_in[8];
    const float* w2    = (const float*)d_in[9];
    const float* b2    = (const float*)d_in[10];
    const float* gg    = (const float*)d_in[11];
    const float* gb    = (const float*)d_in[12];

    float* out = (float*)d_out;
    const size_t BIAS_ELEMS = (size_t)BATCH * SEQ * SEQ;  // 67108864
    float* stats_out   = out + BIAS_ELEMS;       // [gate_impact(4), var(4), entropy(4), contrib(4)]
    float* entropy_out = stats_out + 8;

    // workspace layout (floats)
    float* ws   = (float*)d_ws;
    float* qs   = ws;               // 256
    float* mv   = ws + 256;         // 256
    float* mvw  = ws + 512;         // 256
    float* nrm  = ws + 768;         // 4
    float* Gbuf = ws + 1024;        // 16384
    float* ubuf = ws + 1024 + BATCH * SEQ;  // 16384

    qsum_kernel<<<BATCH, 256, 0, stream>>>(Q, qs);
    small_kernel<<<1, 256, 0, stream>>>(mpv, qs, w1, b1, w2, b2, wmem,
                                        mv, mvw, nrm, entropy_out);
    gu_kernel<<<2 * BATCH, 256, 0, stream>>>(Q, K, mv, mvw, nrm, gg, gb,
                                             Gbuf, ubuf, stats_out);

    const int total_tiles = BATCH * STILES * STILES;   // 262144
    const int waves_per_block = 8;                     // 256 threads, wave32
    bias_kernel<<<total_tiles / waves_per_block, 256, 0, stream>>>(Gbuf, ubuf, amask, out);
}